// CTCLayer_37374805410589
// MI455X (gfx1250) — compile-verified
//
#include <hip/hip_runtime.h>
#include <stdint.h>

// Problem constants (match reference)
#define CTC_B   2048
#define CTC_T   256
#define CTC_L   32
#define CTC_V   128
#define CTC_S   (2 * CTC_L + 1)   // 65 extended states
#define CTC_BLANK (CTC_V - 1)     // 127
#define CTC_EPS 1e-7f
#define CTC_NEG (-1.0e30f)
#define CTC_D   8                 // async prefetch pipeline depth (power of 2)
#define CTC_LN2 0.69314718055994530942f

// Native base-2 exp (v_exp_f32).
#if __has_builtin(__builtin_amdgcn_exp2f)
#define CTC_EXP2(x) __builtin_amdgcn_exp2f(x)
#else
#define CTC_EXP2(x) exp2f(x)
#endif

// Low 32 bits of a generic pointer to __shared__ data == wave-relative LDS byte
// address (flat LDS addresses are {aperture_hi, lds_offset}).
__device__ __forceinline__ unsigned lds_addr32(const void* p) {
    return (unsigned)(uintptr_t)p;
}

// CDNA5 async DMA: 16 bytes/lane global -> LDS, non-temporal (stream-once data;
// 256 MB > 192 MB L2, so don't pollute). Tracked by ASYNCcnt.
__device__ __forceinline__ void async_load_b128_nt(unsigned lds, const void* gaddr) {
    asm volatile("global_load_async_to_lds_b128 %0, %1, off th:TH_LOAD_NT"
                 :
                 : "v"(lds), "v"((unsigned long long)(uintptr_t)gaddr)
                 : "memory");
}

__device__ __forceinline__ void wait_async_le7() {
    asm volatile("s_wait_asynccnt 0x7" ::: "memory");
}
__device__ __forceinline__ void wait_async_le0() {
    asm volatile("s_wait_asynccnt 0x0" ::: "memory");
}

// log2(2^a + 2^b) with -1e30 as -inf surrogate; v_exp/v_log are native base-2.
__device__ __forceinline__ float lae2(float a, float b) {
    float mx = fmaxf(a, b);
    float mn = fminf(a, b);
    return mx + __log2f(1.0f + CTC_EXP2(mn - mx));
}

// One CTC alpha step: read alpha(t-1) from alp[pb], write alpha(t) to alp[wb],
// using the softmax row for time t staged in LDS ring slot rb.
__device__ __forceinline__ void ctc_step(float alp[2][CTC_S + 2],
                                         const float* rowt,
                                         int pb, int wb, int lane,
                                         const int* eidx, const bool* skipb,
                                         const bool* valid) {
    float nv[3];
#pragma unroll
    for (int k = 0; k < 3; ++k) {
        if (!valid[k]) continue;
        const int s = lane + 32 * k;
        float a = lae2(alp[pb][2 + s], alp[pb][1 + s]);  // self + (s-1)
        if (skipb[k]) a = lae2(a, alp[pb][s]);           // + (s-2) if allowed
        nv[k] = a + __log2f(rowt[eidx[k]] + CTC_EPS);
    }
#pragma unroll
    for (int k = 0; k < 3; ++k) {
        if (valid[k]) alp[wb][2 + lane + 32 * k] = nv[k];
    }
    __builtin_amdgcn_wave_barrier();
}

__global__ __launch_bounds__(32)
void ctc_alpha_kernel(const int* __restrict__ y_true,
                      const float* __restrict__ y_pred,
                      float* __restrict__ out) {
    // 8-deep ring of softmax rows (512 B each) + double-buffered alpha with two
    // NEG pad slots in front so s-1 / s-2 shifts are unconditional reads.
    __shared__ float row[CTC_D][CTC_V];
    __shared__ float alp[2][CTC_S + 2];

    const int b    = blockIdx.x;
    const int lane = threadIdx.x;            // wave32: 0..31
    const float* gp  = y_pred + (size_t)b * CTC_T * CTC_V;
    const int*   lab = y_true + b * CTC_L;

    // Per-lane states: s = lane, lane+32, lane+64 (last valid only for lane 0).
    int  eidx[3];
    bool skipb[3];
    bool valid[3];
#pragma unroll
    for (int k = 0; k < 3; ++k) {
        const int s = lane + 32 * k;
        valid[k] = (s < CTC_S);
        int  e  = CTC_BLANK;
        bool sk = false;
        if (valid[k] && (s & 1)) {           // odd state -> a real label
            const int li = s >> 1;
            e  = lab[li];
            sk = (s == 1) || (e != lab[li - 1]);
        }
        eidx[k]  = e;
        skipb[k] = sk;
    }

    if (lane == 0) {
        alp[0][0] = CTC_NEG; alp[0][1] = CTC_NEG;
        alp[1][0] = CTC_NEG; alp[1][1] = CTC_NEG;
    }

    // Prime the pipeline: rows 0..D-1 in flight (1 x b128 per lane = one row).
#pragma unroll
    for (int r = 0; r < CTC_D; ++r) {
        async_load_b128_nt(lds_addr32(&row[r][lane * 4]),
                           gp + (size_t)r * CTC_V + lane * 4);
    }
    wait_async_le7();                        // row 0 landed (in-order completion)
    __builtin_amdgcn_wave_barrier();

    // t = 0: only states 0 and 1 reachable. Alphas kept in log2 domain.
#pragma unroll
    for (int k = 0; k < 3; ++k) {
        const int s = lane + 32 * k;
        if (valid[k]) {
            float v = CTC_NEG;
            if (s < 2) v = __log2f(row[0][eidx[k]] + CTC_EPS);
            alp[0][2 + s] = v;
        }
    }
    __builtin_amdgcn_wave_barrier();

    // Steady state: always one prefetch in, one row consumed; after issuing
    // row t+D-1, asynccnt<=D-1 <=> rows 0..t have landed (in-order completion).
    int t = 1;
#pragma unroll 2
    for (; t <= CTC_T - CTC_D; ++t) {
        const int pf = t + CTC_D - 1;
        async_load_b128_nt(lds_addr32(&row[pf & (CTC_D - 1)][lane * 4]),
                           gp + (size_t)pf * CTC_V + lane * 4);
        wait_async_le7();
        ctc_step(alp, &row[t & (CTC_D - 1)][0],
                 (t & 1) ^ 1, t & 1, lane, eidx, skipb, valid);
    }
    // Tail: all rows already issued; cheap full drain then consume.
#pragma unroll
    for (; t < CTC_T; ++t) {
        wait_async_le0();
        ctc_step(alp, &row[t & (CTC_D - 1)][0],
                 (t & 1) ^ 1, t & 1, lane, eidx, skipb, valid);
    }

    if (lane == 0) {
        const int fb = (CTC_T - 1) & 1;      // buffer holding alpha at t=T-1
        out[b] = -CTC_LN2 * lae2(alp[fb][2 + CTC_S - 1], alp[fb][2 + CTC_S - 2]);
    }
}

extern "C" void kernel_launch(void* const* d_in, const int* in_sizes, int n_in,
                              void* d_out, int out_size, void* d_ws, size_t ws_size,
                              hipStream_t stream) {
    (void)in_sizes; (void)n_in; (void)out_size; (void)d_ws; (void)ws_size;
    const int*   y_true = (const int*)d_in[0];   // [B, L] int32
    const float* y_pred = (const float*)d_in[1]; // [B, T, V] float32 softmax
    float*       out    = (float*)d_out;         // [B, 1] float32

    ctc_alpha_kernel<<<dim3(CTC_B), dim3(32), 0, stream>>>(y_true, y_pred, out);
}